// TemporalAttentionPooler_13477607375140
// MI455X (gfx1250) — compile-verified
//
#include <hip/hip_runtime.h>
#include <stdint.h>

typedef __attribute__((ext_vector_type(2))) float v2f;
typedef __attribute__((ext_vector_type(8))) float v8f;

#define B_ 32
#define T_ 4096
#define D_ 1024
#define S_CHUNKS 16
#define T_CHUNK 256              // T_ / S_CHUNKS
#define TILE_ROWS 16
#define NTILES (T_CHUNK / TILE_ROWS)   // 16
#define THREADS 256
#define NWAVES 8

// dynamic LDS layout (float offsets)
#define TILE_F   (TILE_ROWS * D_)          // 16384
#define TQ_OFF   (2 * TILE_F)              // query: 1024
#define PART_OFF (TQ_OFF + D_)             // per-wave partial scores: 8*16
#define SC_OFF   (PART_OFF + NWAVES * 16)  // final scores: 16
#define SMEM_FLOATS (SC_OFF + 16)          // 33936 floats = 135744 bytes

__global__ void __launch_bounds__(THREADS)
pool_phase1(const float* __restrict__ x, const float* __restrict__ q,
            const unsigned char* __restrict__ mask,
            float* __restrict__ wsm, float* __restrict__ wsl,
            float* __restrict__ wsacc)
{
  extern __shared__ float smem[];
  const int tid   = threadIdx.x;
  const int wave  = tid >> 5;
  const int lane  = tid & 31;
  const int blk   = blockIdx.x;
  const int b     = blk / S_CHUNKS;
  const int chunk = blk % S_CHUNKS;
  const int t0    = chunk * T_CHUNK;

  // stage query into LDS (4 KB)
  {
    float4 v = ((const float4*)q)[tid];
    *(float4*)&smem[TQ_OFF + tid * 4] = v;
  }

  const float* xb = x + (size_t)b * T_ * D_;

  // async-copy one 16x1024 fp32 tile (64 KB) into LDS buffer `buf`
  auto stage = [&](int tile, int buf) {
    uint64_t gbase = (uint64_t)(uintptr_t)(xb + (size_t)(t0 + tile * TILE_ROWS) * D_);
    uint32_t ldsbase = (uint32_t)(uintptr_t)&smem[buf * TILE_F];
#pragma unroll
    for (int k = 0; k < 16; ++k) {           // 16 x b128 per thread -> ASYNCcnt 16/wave
      uint32_t goff = (uint32_t)(k * THREADS + tid) * 16u;  // byte offset
      uint32_t ldsa = ldsbase + goff;
      asm volatile("global_load_async_to_lds_b128 %0, %1, %2"
                   :: "v"(ldsa), "v"(goff), "s"(gbase) : "memory");
    }
  };

  float  m_run = -1e30f, l_run = 0.f;
  float4 acc = {0.f, 0.f, 0.f, 0.f};
  const int d0 = tid * 4;

  stage(0, 0);

  for (int i = 0; i < NTILES; ++i) {
    const int buf = i & 1;
    if (i + 1 < NTILES) {
      stage(i + 1, buf ^ 1);                  // prefetch next tile into other buffer
      asm volatile("s_wait_asynccnt 16" ::: "memory");  // tile i's 16 copies done (in-order)
    } else {
      asm volatile("s_wait_asynccnt 0" ::: "memory");
    }
    __syncthreads();                          // tile i visible to all waves

    const float* lx = &smem[buf * TILE_F];

    // ---- scores via V_WMMA_F32_16X16X4_F32: wave w covers K slice [w*128, w*128+128) ----
    {
      v8f c = {0.f,0.f,0.f,0.f,0.f,0.f,0.f,0.f};
      const int row  = lane & 15;
      const int half = lane >> 4;
#pragma unroll 4
      for (int j = 0; j < 32; ++j) {
        const int dbase = wave * 128 + j * 4 + half * 2;
        v2f a, bq;
        a.x  = lx[row * D_ + dbase];          // A: K0 / K2 per half-wave
        a.y  = lx[row * D_ + dbase + 1];      // A: K1 / K3
        bq.x = smem[TQ_OFF + dbase];          // B: q replicated over N
        bq.y = smem[TQ_OFF + dbase + 1];
        c = __builtin_amdgcn_wmma_f32_16x16x4_f32(
                false, a, false, bq, (short)0, c, false, false);
      }
      // D[m, n=0]: lane 0 holds rows 0..7 in c[0..7], lane 16 holds rows 8..15
      if (lane == 0) {
#pragma unroll
        for (int jj = 0; jj < 8; ++jj) smem[PART_OFF + wave * 16 + jj] = c[jj];
      } else if (lane == 16) {
#pragma unroll
        for (int jj = 0; jj < 8; ++jj) smem[PART_OFF + wave * 16 + 8 + jj] = c[jj];
      }
    }
    __syncthreads();

    // reduce wave partials, scale 1/sqrt(D)=1/32, apply padding mask
    if (tid < 16) {
      float s = 0.f;
#pragma unroll
      for (int w = 0; w < NWAVES; ++w) s += smem[PART_OFF + w * 16 + tid];
      s *= 0.03125f;
      const int t = t0 + i * TILE_ROWS + tid;
      if (mask[(size_t)b * T_ + t]) s = -1e30f;
      smem[SC_OFF + tid] = s;
    }
    __syncthreads();

    // ---- online softmax update + weighted accumulation (4 d's per thread) ----
    float sc[16];
    float tmax = -1e30f;
#pragma unroll
    for (int r = 0; r < 16; ++r) { sc[r] = smem[SC_OFF + r]; tmax = fmaxf(tmax, sc[r]); }
    const float nmax  = fmaxf(m_run, tmax);
    const float scale = __expf(m_run - nmax);
    acc.x *= scale; acc.y *= scale; acc.z *= scale; acc.w *= scale;
    float psum = 0.f;
#pragma unroll
    for (int r = 0; r < 16; ++r) {
      const float p = (sc[r] > -1e29f) ? __expf(sc[r] - nmax) : 0.f;
      psum += p;
      float4 xv = *(const float4*)&lx[r * D_ + d0];
      acc.x += p * xv.x; acc.y += p * xv.y; acc.z += p * xv.z; acc.w += p * xv.w;
    }
    l_run = l_run * scale + psum;
    m_run = nmax;
    __syncthreads();                          // all reads of buf done before re-stage
  }

  if (tid == 0) { wsm[blk] = m_run; wsl[blk] = l_run; }
  *(float4*)&wsacc[(size_t)blk * D_ + d0] = acc;
}

__global__ void __launch_bounds__(THREADS)
pool_phase2(const float* __restrict__ wsm, const float* __restrict__ wsl,
            const float* __restrict__ wsacc, float* __restrict__ out)
{
  const int tid = threadIdx.x;
  const int b   = blockIdx.x;

  float m[S_CHUNKS], l[S_CHUNKS], e[S_CHUNKS];
  float M = -1e30f;
#pragma unroll
  for (int s = 0; s < S_CHUNKS; ++s) {
    m[s] = wsm[b * S_CHUNKS + s];
    l[s] = wsl[b * S_CHUNKS + s];
    M = fmaxf(M, m[s]);
  }
  float L = 0.f;
#pragma unroll
  for (int s = 0; s < S_CHUNKS; ++s) {
    e[s] = (m[s] > -1e29f) ? __expf(m[s] - M) : 0.f;
    L += l[s] * e[s];
  }
  const float inv = 1.f / L;                  // lengths >= 1 guarantees L > 0
  const int d0 = tid * 4;
  float4 a = {0.f, 0.f, 0.f, 0.f};
#pragma unroll
  for (int s = 0; s < S_CHUNKS; ++s) {
    float4 v = *(const float4*)&wsacc[(size_t)(b * S_CHUNKS + s) * D_ + d0];
    a.x += e[s] * v.x; a.y += e[s] * v.y; a.z += e[s] * v.z; a.w += e[s] * v.w;
  }
  a.x *= inv; a.y *= inv; a.z *= inv; a.w *= inv;
  *(float4*)&out[(size_t)b * D_ + d0] = a;
}

extern "C" void kernel_launch(void* const* d_in, const int* in_sizes, int n_in,
                              void* d_out, int out_size, void* d_ws, size_t ws_size,
                              hipStream_t stream) {
  const float*         x    = (const float*)d_in[0];
  const float*         q    = (const float*)d_in[1];
  const unsigned char* mask = (const unsigned char*)d_in[2];   // jnp bool -> 1 byte
  float* out = (float*)d_out;

  float* wsm   = (float*)d_ws;                 // [B*S] running max
  float* wsl   = wsm + B_ * S_CHUNKS;          // [B*S] running denom
  float* wsacc = wsl + B_ * S_CHUNKS;          // [B*S, D] partial weighted sums (16B aligned)

  const size_t smem_bytes = SMEM_FLOATS * sizeof(float);  // 135744 B (2 WGs / 320KB WGP)
  pool_phase1<<<dim3(B_ * S_CHUNKS), THREADS, smem_bytes, stream>>>(x, q, mask, wsm, wsl, wsacc);
  pool_phase2<<<dim3(B_), THREADS, 0, stream>>>(wsm, wsl, wsacc, out);
}